// HoughVoting_10393820857096
// MI455X (gfx1250) — compile-verified
//
#include <hip/hip_runtime.h>

#define NCLS 22
#define SKIPF 8
#define IMG_H 480
#define IMG_W 640
#define HS 60
#define WS 80
#define PPTS 4800      // HS*WS
#define NTILE 300      // PPTS/16

typedef float v2f __attribute__((ext_vector_type(2)));
typedef float v8f __attribute__((ext_vector_type(8)));

// ---------------------------------------------------------------- utilities
__global__ void k_zero(unsigned int* counts, int n) {
    int t = blockIdx.x * blockDim.x + threadIdx.x;
    if (t < n) counts[t] = 0u;
}

// ---------------------------------------------------------------- gather
// Build compact per-point arrays:
//   A4[n][p] = (u, v, -(u*px+v*py), 0)         -> WMMA A operand rows (K=4)
//   B4[p]    = (px, py, 1, 0)                  -> WMMA B operand cols (K=4)
//   E4[n][p] = (px, py, 0.9*nrm, bitcast(lab)) -> epilogue per-voter data
__global__ void k_gather(const int* __restrict__ label, const float* __restrict__ vp,
                         float4* __restrict__ A4, float4* __restrict__ E4,
                         float4* __restrict__ B4, float* __restrict__ z_s,
                         unsigned int* __restrict__ counts, int N) {
    int tid = blockIdx.x * blockDim.x + threadIdx.x;
    if (tid >= N * PPTS) return;
    int n = tid / PPTS, p = tid - n * PPTS;
    int wx = p % WS, hy = p / WS;
    int gx = wx * SKIPF, gy = hy * SKIPF;
    int lab = label[((long)n * IMG_H + gy) * IMG_W + gx];
    long hw = (long)IMG_H * IMG_W;
    long base = ((long)n * 3 * NCLS + 3 * lab) * hw + (long)gy * IMG_W + gx;
    float u = vp[base], v = vp[base + hw], z = vp[base + 2 * hw];
    float px = (float)gx, py = (float)gy;
    float nrm = sqrtf(u * u + v * v) + 1e-8f;
    A4[tid] = make_float4(u, v, -(u * px + v * py), 0.f);
    E4[tid] = make_float4(px, py, 0.9f * nrm, __int_as_float(lab));
    z_s[tid] = z;
    if (n == 0) B4[p] = make_float4(px, py, 1.f, 0.f);
    atomicAdd(&counts[n * NCLS + lab], 1u);
}

// ---------------------------------------------------------------- voting
// One block per (n, 16-column j-block); 8 waves split the i-tiles.
// Numerator tile via V_WMMA_F32_16X16X4_F32; native v_sqrt_f32 dist in
// epilogue; per-class votes accumulated in LDS with integer atomics.
__global__ __launch_bounds__(256) void k_vote(
        const float4* __restrict__ A4, const float4* __restrict__ E4,
        const float4* __restrict__ B4, unsigned int* __restrict__ hough) {
    __shared__ unsigned int hsh[NCLS * 16];
    int blk = blockIdx.x;
    int n = blk / NTILE;
    int j0 = (blk - n * NTILE) * 16;
    int t = threadIdx.x;
    for (int s = t; s < NCLS * 16; s += 256) hsh[s] = 0u;
    __syncthreads();

    int wave = t >> 5;
    int lane = t & 31;
    int half = lane >> 4;   // lane half selects K-pair (2h, 2h+1)
    int ml   = lane & 15;   // A-row index / B-column index within tile
    int j = j0 + ml;
    float pxj = (float)((j % WS) * SKIPF);
    float pyj = (float)((j / WS) * SKIPF);

    // B operand: lane holds B[k=2h..2h+1][col=ml]; fixed for this block.
    const float2* bp = (const float2*)&B4[j];
    float2 bl = bp[half];
    v2f b; b[0] = bl.x; b[1] = bl.y;

    const float4* A4n = A4 + (long)n * PPTS;
    const float4* E4n = E4 + (long)n * PPTS;

    // software pipeline: preload first A operand
    float2 al = ((const float2*)&A4n[wave * 16 + ml])[half];

    for (int it = wave; it < NTILE; it += 8) {
        int i0 = it * 16;
        v2f a; a[0] = al.x; a[1] = al.y;
        v8f cz = {};
        // num[i][j] = u_i*px_j + v_i*py_j - (u_i*px_i + v_i*py_i)
        v8f d = __builtin_amdgcn_wmma_f32_16x16x4_f32(
            false, a, false, b, (short)0, cz, false, false);
        // issue next tile's A load; hides under the 8-row epilogue
        int itn = (it + 8 < NTILE) ? (it + 8) : it;
        al = ((const float2*)&A4n[itn * 16 + ml])[half];

        // batch-issue all 8 per-row loads (uniform per half-wave ->
        // broadcast); overlaps 7 loads' latency with the compare math
        float4 er[8];
#pragma unroll
        for (int r = 0; r < 8; ++r) er[r] = E4n[i0 + 8 * half + r];
#pragma unroll
        for (int r = 0; r < 8; ++r) {
            float4 e = er[r];
            float dx = pxj - e.x, dy = pyj - e.y;
            float d2 = dx * dx + dy * dy;
            float dist = __builtin_amdgcn_sqrtf(d2);  // raw v_sqrt_f32
            int labi = __float_as_int(e.w);
            bool vote = (d[r] >= e.z * (dist + 1e-8f)) &&
                        (d2 > 0.f) && (labi > 0);
            if (vote) atomicAdd(&hsh[labi * 16 + ml], 1u);
        }
    }
    __syncthreads();
    for (int s = t; s < NCLS * 16; s += 256) {
        int cc = s >> 4, jl = s & 15;
        hough[((long)(n * NCLS + cc)) * PPTS + j0 + jl] = hsh[s];
    }
}

// ---------------------------------------------------------------- finalize
// One block per (n, c): first-occurrence argmax over hough row, then
// deterministic recompute of z-mean for the winning column only.
__global__ __launch_bounds__(256) void k_final(
        const unsigned int* __restrict__ hough, const unsigned int* __restrict__ counts,
        const float4* __restrict__ A4, const float4* __restrict__ E4,
        const float* __restrict__ z_s, const float* __restrict__ extents,
        const float* __restrict__ meta, float* __restrict__ out, int N) {
    int n = blockIdx.x / NCLS, c = blockIdx.x % NCLS;
    int t = threadIdx.x;
    int ob = (n * NCLS + c) * 7;
    int op = N * NCLS * 7 + ob;
    if (c == 0) {                       // class 0 never valid -> zeros
        if (t < 7) { out[ob + t] = 0.f; out[op + t] = 0.f; }
        return;
    }
    __shared__ unsigned int smax[256];
    __shared__ int sidx[256];
    __shared__ float sza[256];
    __shared__ float sna[256];

    const unsigned int* hrow = hough + ((long)(n * NCLS + c)) * PPTS;
    unsigned int bv = 0u; int bj = 0x7fffffff;
    for (int j = t; j < PPTS; j += 256) {
        unsigned int hv = hrow[j];
        if (hv > bv || (hv == bv && j < bj)) { bv = hv; bj = j; }
    }
    smax[t] = bv; sidx[t] = bj;
    __syncthreads();
    for (int s = 128; s > 0; s >>= 1) {
        if (t < s) {
            unsigned int ov = smax[t + s]; int oj = sidx[t + s];
            if (ov > smax[t] || (ov == smax[t] && oj < sidx[t])) {
                smax[t] = ov; sidx[t] = oj;
            }
        }
        __syncthreads();
    }
    unsigned int mvu = smax[0];
    int jb = sidx[0];
    float pxj = (float)((jb % WS) * SKIPF);
    float pyj = (float)((jb / WS) * SKIPF);

    const float4* A4n = A4 + (long)n * PPTS;
    const float4* E4n = E4 + (long)n * PPTS;
    const float* zn = z_s + (long)n * PPTS;
    float lz = 0.f, ln = 0.f;
    for (int i = t; i < PPTS; i += 256) {
        float4 e = E4n[i];
        if (__float_as_int(e.w) != c) continue;
        float4 a = A4n[i];
        float dx = pxj - e.x, dy = pyj - e.y;
        float d2 = dx * dx + dy * dy;
        float dist = __builtin_amdgcn_sqrtf(d2);  // same op as k_vote
        float num = a.x * pxj + a.y * pyj + a.z;
        if (num >= e.z * (dist + 1e-8f) && d2 > 0.f) {
            ln += 1.f; lz += zn[i];
        }
    }
    sza[t] = lz; sna[t] = ln;
    __syncthreads();
    for (int s = 128; s > 0; s >>= 1) {
        if (t < s) { sza[t] += sza[t + s]; sna[t] += sna[t + s]; }
        __syncthreads();
    }
    if (t == 0) {
        float mv = (float)mvu;
        float cnt = (float)counts[n * NCLS + c];
        bool valid = (cnt * 64.f >= 500.f) && (mv >= 10.f) && (mv >= 0.02f * cnt);
        float vm = valid ? 1.f : 0.f;
        float nin = sna[0], zsum = sza[0];
        float zmean = zsum / fmaxf(nin, 1.f);
        float zc = fmaxf(zmean, 0.001f);
        float fx = meta[n * 9 + 0], ppx = meta[n * 9 + 2];
        float fy = meta[n * 9 + 4], ppy = meta[n * 9 + 5];
        float bw = extents[c * 3 + 0] * fx / zc;
        float bh = extents[c * 3 + 1] * fy / zc;
        float cx = pxj, cy = pyj;
        out[ob + 0] = (float)n * vm;
        out[ob + 1] = (float)c * vm;
        out[ob + 2] = (cx - bw * 0.5f) * vm;
        out[ob + 3] = (cy - bh * 0.5f) * vm;
        out[ob + 4] = (cx + bw * 0.5f) * vm;
        out[ob + 5] = (cy + bh * 0.5f) * vm;
        out[ob + 6] = mv * vm;
        out[op + 0] = vm;
        out[op + 1] = 0.f;
        out[op + 2] = 0.f;
        out[op + 3] = 0.f;
        out[op + 4] = (cx - ppx) * zc / fx * vm;
        out[op + 5] = (cy - ppy) * zc / fy * vm;
        out[op + 6] = zc * vm;
    }
}

// ---------------------------------------------------------------- launch
extern "C" void kernel_launch(void* const* d_in, const int* in_sizes, int n_in,
                              void* d_out, int out_size, void* d_ws, size_t ws_size,
                              hipStream_t stream) {
    (void)n_in; (void)out_size; (void)ws_size;
    const int*   label   = (const int*)d_in[0];
    const float* vp      = (const float*)d_in[1];
    const float* extents = (const float*)d_in[2];
    const float* meta    = (const float*)d_in[4];   // d_in[3] (poses) unused
    float* out = (float*)d_out;
    int N = in_sizes[0] / (IMG_H * IMG_W);

    char* ws = (char*)d_ws;
    size_t o = 0;
    float4* A4 = (float4*)(ws + o); o += (size_t)N * PPTS * sizeof(float4);
    float4* E4 = (float4*)(ws + o); o += (size_t)N * PPTS * sizeof(float4);
    float4* B4 = (float4*)(ws + o); o += (size_t)PPTS * sizeof(float4);
    float*  z_s = (float*)(ws + o); o += (size_t)N * PPTS * sizeof(float);
    unsigned int* counts = (unsigned int*)(ws + o);
    o += ((size_t)N * NCLS * 4 + 255) & ~(size_t)255;
    unsigned int* hough = (unsigned int*)(ws + o);

    k_zero<<<1, 64, 0, stream>>>(counts, N * NCLS);
    int tot = N * PPTS;
    k_gather<<<(tot + 255) / 256, 256, 0, stream>>>(label, vp, A4, E4, B4, z_s, counts, N);
    k_vote<<<N * NTILE, 256, 0, stream>>>(A4, E4, B4, hough);
    k_final<<<N * NCLS, 256, 0, stream>>>(hough, counts, A4, E4, z_s, extents, meta, out, N);
}